// Attention_88313117540783
// MI455X (gfx1250) — compile-verified
//
#include <hip/hip_runtime.h>

typedef float v2f __attribute__((ext_vector_type(2)));
typedef float v4f __attribute__((ext_vector_type(4)));
typedef float v8f __attribute__((ext_vector_type(8)));

#define LOG2E 1.4426950408889634f

// ---------------------------------------------------------------------------
// GEMM + bias: Y[M,N] = X[M,H] @ W[H,N] + bias[N], fp32 via v_wmma_f32_16x16x4_f32.
// One wave per 16x16 output tile; K advances by 4 per WMMA.
// A-frag layout (16x4 f32): lane = {l15 = M, half = K-pair}; a.x=K0+2h, a.y=K0+2h+1
// B-frag layout (4x16 f32): lane = {l15 = N, half = K-pair}; b.x=W[K0+2h][n], b.y=W[K0+2h+1][n]
// C/D layout: VGPR j -> (M = tm + half*8 + j, N = tn + l15)
// ---------------------------------------------------------------------------
__global__ __launch_bounds__(256) void gemm_bias_wmma_f32(
    const float* __restrict__ X, const float* __restrict__ W,
    const float* __restrict__ bias, float* __restrict__ Y,
    int M, int H, int N) {
  int gtid  = blockIdx.x * blockDim.x + threadIdx.x;
  int wave  = gtid >> 5;
  int lane  = threadIdx.x & 31;
  int tilesN = N >> 4;
  int tm = (wave / tilesN) << 4;   // wave-uniform
  int tn = (wave % tilesN) << 4;
  if (tm >= M) return;             // uniform per wave -> EXEC stays all-ones

  int half = lane >> 4;            // 0 or 1 (K-pair select)
  int l15  = lane & 15;

  const float* arow = X + (size_t)(tm + l15) * H + 2 * half;
  const float* bcol = W + (size_t)(2 * half) * N + tn + l15;

  v8f acc = {};
#pragma unroll 4
  for (int k = 0; k < H; k += 4) {
    v2f a, b;
    a.x = arow[k];
    a.y = arow[k + 1];
    b.x = bcol[(size_t)k * N];
    b.y = bcol[(size_t)(k + 1) * N];
    acc = __builtin_amdgcn_wmma_f32_16x16x4_f32(
        /*neg_a=*/false, a, /*neg_b=*/false, b,
        /*c_mod=*/(short)0, acc, /*reuse_a=*/false, /*reuse_b=*/false);
  }

  int   col   = tn + l15;
  float bb    = bias[col];
  int   mbase = tm + half * 8;
#pragma unroll
  for (int j = 0; j < 8; ++j)
    Y[(size_t)(mbase + j) * N + col] = acc[j] + bb;
}

// ---------------------------------------------------------------------------
// Fused energy + v-projection + softmax.
// One block per (b,t); 256 threads each own 4 of the S=1024 scores.
// score[s] = sum_a tanh(eh[b,s,a] + dh[b,t,a]) * Wv[a] + bv, then softmax over s.
// dh row + Wv staged in LDS; eh streamed with 16B loads (L2-resident, 4 MB).
// tanh via hardware v_exp_f32 + v_rcp_f32.
// ---------------------------------------------------------------------------
__device__ __forceinline__ float fast_tanh(float x) {
  x = fminf(fmaxf(x, -10.0f), 10.0f);               // keep exp2 finite
  float e = __builtin_amdgcn_exp2f(x * (2.0f * LOG2E)); // e^(2x)
  return (e - 1.0f) * __builtin_amdgcn_rcpf(e + 1.0f);
}

__global__ __launch_bounds__(256) void energy_softmax(
    const float* __restrict__ eh,   // [B,S,A]
    const float* __restrict__ dh,   // [B,T,A]
    const float* __restrict__ Wv,   // [A]
    const float* __restrict__ bv,   // [1]
    float* __restrict__ out) {      // [B,T,S]
  const int S = 1024, A = 256;      // T = 256 (bt>>8 gives b)
  int bt  = blockIdx.x;
  int tid = threadIdx.x;

  __shared__ float s_d[A];
  __shared__ float s_w[A];
  __shared__ float red[256];

  s_d[tid] = dh[(size_t)bt * A + tid];
  s_w[tid] = Wv[tid];
  __syncthreads();

  const v4f* sd4 = (const v4f*)s_d;
  const v4f* sw4 = (const v4f*)s_w;
  float bvv = bv[0];

  float sc[4];
#pragma unroll
  for (int i = 0; i < 4; ++i) {
    int s = i * 256 + tid;
    const v4f* ep = (const v4f*)(eh + ((size_t)(bt >> 8) * S + s) * A);
    float acc = 0.0f;
#pragma unroll 4
    for (int a4 = 0; a4 < A / 4; ++a4) {
      v4f e = ep[a4];
      v4f d = sd4[a4];
      v4f w = sw4[a4];
      acc += fast_tanh(e.x + d.x) * w.x;
      acc += fast_tanh(e.y + d.y) * w.y;
      acc += fast_tanh(e.z + d.z) * w.z;
      acc += fast_tanh(e.w + d.w) * w.w;
    }
    sc[i] = acc + bvv;
  }

  // block max
  float m = fmaxf(fmaxf(sc[0], sc[1]), fmaxf(sc[2], sc[3]));
  red[tid] = m;
  __syncthreads();
  for (int off = 128; off > 0; off >>= 1) {
    if (tid < off) red[tid] = fmaxf(red[tid], red[tid + off]);
    __syncthreads();
  }
  float bm = red[0];
  __syncthreads();

  // exp + block sum
  float p[4];
  float sum = 0.0f;
#pragma unroll
  for (int i = 0; i < 4; ++i) {
    p[i] = __builtin_amdgcn_exp2f((sc[i] - bm) * LOG2E);
    sum += p[i];
  }
  red[tid] = sum;
  __syncthreads();
  for (int off = 128; off > 0; off >>= 1) {
    if (tid < off) red[tid] += red[tid + off];
    __syncthreads();
  }
  float rinv = __builtin_amdgcn_rcpf(red[0]);

#pragma unroll
  for (int i = 0; i < 4; ++i)
    out[(size_t)bt * S + i * 256 + tid] = p[i] * rinv;
}

// ---------------------------------------------------------------------------
extern "C" void kernel_launch(void* const* d_in, const int* in_sizes, int n_in,
                              void* d_out, int out_size, void* d_ws, size_t ws_size,
                              hipStream_t stream) {
  const int B = 4, S = 1024, T = 256, H = 512, A = 256;

  const float* enc = (const float*)d_in[0];  // [B,S,H]
  const float* dec = (const float*)d_in[1];  // [B,T,H]
  const float* Wh  = (const float*)d_in[2];  // [H,A]
  const float* bh  = (const float*)d_in[3];  // [A]
  const float* Ws  = (const float*)d_in[4];  // [H,A]
  const float* bs  = (const float*)d_in[5];  // [A]
  const float* Wv  = (const float*)d_in[6];  // [A,1]
  const float* bv  = (const float*)d_in[7];  // [1]
  float* out = (float*)d_out;

  float* eh  = (float*)d_ws;                    // [B,S,A] = 4 MB
  float* dhp = eh + (size_t)B * S * A;          // [B,T,A] = 1 MB

  int Me = B * S;                               // 4096
  int Md = B * T;                               // 1024
  int waves_e = (Me / 16) * (A / 16);           // 4096 waves
  int waves_d = (Md / 16) * (A / 16);           // 1024 waves

  gemm_bias_wmma_f32<<<waves_e / 8, 256, 0, stream>>>(enc, Wh, bh, eh, Me, H, A);
  gemm_bias_wmma_f32<<<waves_d / 8, 256, 0, stream>>>(dec, Ws, bs, dhp, Md, H, A);
  energy_softmax<<<B * T, 256, 0, stream>>>(eh, dhp, Wv, bv, out);
}